// GAT_86672440033563
// MI455X (gfx1250) — compile-verified
//
#include <hip/hip_runtime.h>
#include <math.h>

typedef __attribute__((ext_vector_type(2))) float v2f;
typedef __attribute__((ext_vector_type(8))) float v8f;

#define NEG_SLOPE 0.2f

// ---------------------------------------------------------------- utilities
__global__ void fill_kernel(float* __restrict__ p, float v, int n) {
    int i = blockIdx.x * blockDim.x + threadIdx.x;
    if (i < n) p[i] = v;
}

__device__ __forceinline__ void edge_nodes(const int* __restrict__ ei, int E,
                                           int e, int& src, int& dst) {
    if (e < E) { src = ei[e]; dst = ei[E + e]; }
    else       { src = dst = e - E; }           // self loops appended
}

// ---------------------------------------------------------------- WMMA GEMM
// C[nrows x NCOL] = A[nrows x K] * B[K x NCOL], all fp32, via
// V_WMMA_F32_16X16X4_F32.  One wave per 16x16 C tile; NCOL/16 waves per block.
// nrows must be a multiple of 16 (100000 = 6250*16).  No divergence: EXEC
// stays all-ones as WMMA requires.
template<int K, int NCOL>
__global__ void wmma_gemm_f32(const float* __restrict__ A,
                              const float* __restrict__ B,
                              float* __restrict__ C) {
    const int lane = threadIdx.x & 31;
    const int wave = threadIdx.x >> 5;
    const int lm   = lane & 15;        // M (A) / N (B,C) position
    const int lh   = lane >> 4;        // half-wave selector
    const int rowBase = blockIdx.x << 4;
    const int colBase = wave << 4;

    const float* arow = A + (size_t)(rowBase + lm) * K;     // A row-major
    const float* bcol = B + (size_t)colBase + lm;           // B row-major

    v8f acc = {0.f, 0.f, 0.f, 0.f, 0.f, 0.f, 0.f, 0.f};

    // ISA A-layout 16x4: v0 holds K=k0+2*lh, v1 holds K=k0+1+2*lh (contiguous
    // pair -> one 8B load).  B-layout 4x16: v0 row K=k0+2*lh, v1 row +1.
#pragma unroll 8
    for (int k0 = 0; k0 < K; k0 += 4) {
        const int ka = k0 + 2 * lh;
        v2f a = *(const v2f*)(arow + ka);
        v2f b;
        b.x = bcol[(size_t)ka * NCOL];
        b.y = bcol[(size_t)(ka + 1) * NCOL];
        acc = __builtin_amdgcn_wmma_f32_16x16x4_f32(
            /*neg_a=*/false, a, /*neg_b=*/false, b,
            /*c_mod=*/(short)0, acc, /*reuse_a=*/false, /*reuse_b=*/false);
    }

    // C/D layout: VGPR r, lane l -> row = r + 8*lh, col = lm
    float* crow = C + (size_t)(rowBase + 8 * lh) * NCOL + colBase + lm;
#pragma unroll
    for (int r = 0; r < 8; ++r)
        crow[(size_t)r * NCOL] = acc[r];
}

// ------------------------------------------------- per-node attention scores
// h is [n, H, C] contiguous; idx = node*H + head
template<int H, int C>
__global__ void scores_kernel(const float* __restrict__ h,
                              const float* __restrict__ a_src,
                              const float* __restrict__ a_dst,
                              float* __restrict__ s_src,
                              float* __restrict__ s_dst, int n) {
    int idx = blockIdx.x * blockDim.x + threadIdx.x;
    if (idx >= n * H) return;
    const int hh = idx % H;
    const float* hp = h + (size_t)idx * C;
    float ss = 0.f, sd = 0.f;
#pragma unroll
    for (int c = 0; c < C; ++c) {
        float v = hp[c];
        ss += v * a_src[hh * C + c];
        sd += v * a_dst[hh * C + c];
    }
    s_src[idx] = ss;
    s_dst[idx] = sd;
}

// ---------------------------------------------------- segment-softmax passes
template<int H>
__global__ void edge_max_kernel(const int* __restrict__ ei, int E, int Etot,
                                const float* __restrict__ s_src,
                                const float* __restrict__ s_dst,
                                float* __restrict__ m) {
    int e = blockIdx.x * blockDim.x + threadIdx.x;
    if (e >= Etot) return;
    int src, dst; edge_nodes(ei, E, e, src, dst);
#pragma unroll
    for (int h = 0; h < H; ++h) {
        float v = s_src[src * H + h] + s_dst[dst * H + h];
        v = v > 0.f ? v : NEG_SLOPE * v;                 // leaky_relu
        atomicMax(&m[dst * H + h], v);                   // global_atomic_max_num_f32
    }
}

template<int H>
__global__ void edge_expsum_kernel(const int* __restrict__ ei, int E, int Etot,
                                   const float* __restrict__ s_src,
                                   const float* __restrict__ s_dst,
                                   const float* __restrict__ m,
                                   float* __restrict__ denom) {
    int e = blockIdx.x * blockDim.x + threadIdx.x;
    if (e >= Etot) return;
    int src, dst; edge_nodes(ei, E, e, src, dst);
#pragma unroll
    for (int h = 0; h < H; ++h) {
        float v = s_src[src * H + h] + s_dst[dst * H + h];
        v = v > 0.f ? v : NEG_SLOPE * v;
        atomicAdd(&denom[dst * H + h], expf(v - m[dst * H + h]));
    }
}

// -------------------------------------------- weighted message aggregation
// One wave per edge; lanes sweep the H*C features (coalesced gather of
// h[src], atomic scatter-add into out[dst]).  Alpha recomputed per head
// (4 cached loads) instead of materializing 27MB of per-edge alphas.
template<int H, int C>
__global__ void edge_aggr_kernel(const int* __restrict__ ei, int E, int Etot,
                                 const float* __restrict__ s_src,
                                 const float* __restrict__ s_dst,
                                 const float* __restrict__ m,
                                 const float* __restrict__ denom,
                                 const float* __restrict__ hfeat,
                                 float* __restrict__ out) {
    constexpr int F   = H * C;
    constexpr int VEC = F / 32;
    const int lane = threadIdx.x & 31;
    const int e = (blockIdx.x * blockDim.x + threadIdx.x) >> 5;
    if (e >= Etot) return;
    int src, dst; edge_nodes(ei, E, e, src, dst);
    const float* hs = hfeat + (size_t)src * F;
    float*       od = out   + (size_t)dst * F;
#pragma unroll
    for (int v = 0; v < VEC; ++v) {
        const int f = v * 32 + lane;
        const int h = f / C;                              // uniform per v when C==32
        float ev = s_src[src * H + h] + s_dst[dst * H + h];
        ev = ev > 0.f ? ev : NEG_SLOPE * ev;
        const float alpha = expf(ev - m[dst * H + h]) / denom[dst * H + h];
        atomicAdd(&od[f], alpha * hs[f]);
    }
}

// ------------------------------------------------------------ elementwise
__global__ void elu_bias_kernel(float* __restrict__ x,
                                const float* __restrict__ b, int n, int F) {
    int i = blockIdx.x * blockDim.x + threadIdx.x;
    if (i >= n) return;
    float t = x[i] + b[i % F];
    x[i] = t > 0.f ? t : expm1f(t);                       // elu(alpha=1)
}

__global__ void bias_add_kernel(float* __restrict__ x,
                                const float* __restrict__ b, int n, int F) {
    int i = blockIdx.x * blockDim.x + threadIdx.x;
    if (i >= n) return;
    x[i] += b[i % F];
}

// ----------------------------------------------------------------- driver
extern "C" void kernel_launch(void* const* d_in, const int* in_sizes, int n_in,
                              void* d_out, int out_size, void* d_ws, size_t ws_size,
                              hipStream_t stream) {
    (void)n_in; (void)out_size; (void)ws_size;
    const float* x      = (const float*)d_in[0];
    const int*   ei     = (const int*)  d_in[1];
    const float* W1     = (const float*)d_in[2];
    const float* a_src1 = (const float*)d_in[3];
    const float* a_dst1 = (const float*)d_in[4];
    const float* b1     = (const float*)d_in[5];
    const float* W2     = (const float*)d_in[6];
    const float* a_src2 = (const float*)d_in[7];
    const float* a_dst2 = (const float*)d_in[8];
    const float* b2     = (const float*)d_in[9];
    float* out = (float*)d_out;

    const int N    = in_sizes[0] / 128;   // 100000
    const int E    = in_sizes[1] / 2;     // 1600000
    const int Etot = E + N;               // + self loops

    // workspace layout (floats), ~123 MB total
    float* ws    = (float*)d_ws;
    float* h1    = ws; ws += (size_t)N * 128;
    float* ssrc1 = ws; ws += (size_t)N * 4;
    float* sdst1 = ws; ws += (size_t)N * 4;
    float* m1    = ws; ws += (size_t)N * 4;
    float* dn1   = ws; ws += (size_t)N * 4;
    float* out1  = ws; ws += (size_t)N * 128;
    float* h2    = ws; ws += (size_t)N * 32;
    float* ssrc2 = ws; ws += (size_t)N;
    float* sdst2 = ws; ws += (size_t)N;
    float* m2    = ws; ws += (size_t)N;
    float* dn2   = ws; ws += (size_t)N;

    auto cdiv = [](int a, int b) { return (a + b - 1) / b; };
    const float NEGINF = -INFINITY;

    // re-init accumulators every call (deterministic under graph replay)
    fill_kernel<<<cdiv(N * 4, 256), 256, 0, stream>>>(m1, NEGINF, N * 4);
    fill_kernel<<<cdiv(N * 4, 256), 256, 0, stream>>>(dn1, 0.f, N * 4);
    fill_kernel<<<cdiv(N * 128, 256), 256, 0, stream>>>(out1, 0.f, N * 128);
    fill_kernel<<<cdiv(N, 256), 256, 0, stream>>>(m2, NEGINF, N);
    fill_kernel<<<cdiv(N, 256), 256, 0, stream>>>(dn2, 0.f, N);
    fill_kernel<<<cdiv(N * 32, 256), 256, 0, stream>>>(out, 0.f, N * 32);

    // ----- layer 1 (H=4, C=32, concat) -----
    wmma_gemm_f32<128, 128><<<N / 16, 256, 0, stream>>>(x, W1, h1);
    scores_kernel<4, 32><<<cdiv(N * 4, 256), 256, 0, stream>>>(h1, a_src1, a_dst1,
                                                               ssrc1, sdst1, N);
    edge_max_kernel<4><<<cdiv(Etot, 256), 256, 0, stream>>>(ei, E, Etot, ssrc1, sdst1, m1);
    edge_expsum_kernel<4><<<cdiv(Etot, 256), 256, 0, stream>>>(ei, E, Etot, ssrc1, sdst1,
                                                               m1, dn1);
    edge_aggr_kernel<4, 32><<<cdiv(Etot, 8), 256, 0, stream>>>(ei, E, Etot, ssrc1, sdst1,
                                                               m1, dn1, h1, out1);
    elu_bias_kernel<<<cdiv(N * 128, 256), 256, 0, stream>>>(out1, b1, N * 128, 128);

    // ----- layer 2 (H=1, C=32, mean over 1 head == identity) -----
    wmma_gemm_f32<128, 32><<<N / 16, 64, 0, stream>>>(out1, W2, h2);
    scores_kernel<1, 32><<<cdiv(N, 256), 256, 0, stream>>>(h2, a_src2, a_dst2,
                                                           ssrc2, sdst2, N);
    edge_max_kernel<1><<<cdiv(Etot, 256), 256, 0, stream>>>(ei, E, Etot, ssrc2, sdst2, m2);
    edge_expsum_kernel<1><<<cdiv(Etot, 256), 256, 0, stream>>>(ei, E, Etot, ssrc2, sdst2,
                                                               m2, dn2);
    edge_aggr_kernel<1, 32><<<cdiv(Etot, 8), 256, 0, stream>>>(ei, E, Etot, ssrc2, sdst2,
                                                               m2, dn2, h2, out);
    bias_add_kernel<<<cdiv(N * 32, 256), 256, 0, stream>>>(out, b2, N * 32, 32);
}